// FreTransferInv_28441273434573
// MI455X (gfx1250) — compile-verified
//
#include <hip/hip_runtime.h>
#include <cstdint>
#include <cstddef>

// ---------------------------------------------------------------------------
// Inverse Haar synthesis (conv_transpose2d, stride 2, block-diagonal 2x2
// filters). Shapes: x (8,16,512,512) f32 -> out (8,4,1024,1024) f32.
// Memory-bound: 256 MiB total traffic @ 23.3 TB/s ~= 11 us. Strategy:
// one TDM tensor_load_to_lds stages the 4-channel x 2-row x 512 input tile
// (16 KB) into LDS per block; threads butterfly from LDS with ds_load_b128
// and write coalesced, NON-TEMPORAL global_store_b128 (streaming output
// should not retain dirty lines in L2).
// ---------------------------------------------------------------------------

#if defined(__HIP_DEVICE_COMPILE__) && defined(__gfx1250__)
#  if __has_builtin(__builtin_amdgcn_tensor_load_to_lds)
#    define FTI_HAS_TDM 1
#    if __has_include(<hip/amd_detail/amd_gfx1250_TDM.h>)
#      define FTI_TDM_ARITY 6   /* amdgpu-toolchain / clang-23 flavor */
#    else
#      define FTI_TDM_ARITY 5   /* ROCm 7.2 / clang-22 flavor (confirmed) */
#    endif
#  endif
#endif

typedef unsigned int fti_u32x4 __attribute__((ext_vector_type(4)));
typedef int          fti_i32x8 __attribute__((ext_vector_type(8)));
typedef int          fti_i32x4 __attribute__((ext_vector_type(4)));
typedef float        fti_f32x4 __attribute__((ext_vector_type(4)));

__global__ __launch_bounds__(256) void fti_haar_inv_kernel(
    const float* __restrict__ x,
    const float* __restrict__ w1,
    const float* __restrict__ w2,
    float* __restrict__ out)
{
    // Block decomposition: blockIdx.x = b*1024 + j*256 + hb
    //   hb in [0,256): pair of input rows {2hb, 2hb+1}
    //   j  in [0,4)  : output channel
    //   b  in [0,8)  : batch
    const int hb = (int)(blockIdx.x & 255u);
    const int j  = (int)((blockIdx.x >> 8) & 3u);
    const int b  = (int)(blockIdx.x >> 10);

    // Synthesis column/row filter taps (Haar: p0=p1=q0=s, q1=-s, s=1/sqrt2)
    const float p0 = w1[0], p1 = w1[1];
    const float q0 = w2[0], q1 = w2[1];

    const int hh = (int)(threadIdx.x >> 7);   // which of the two input rows
    const int wq = (int)(threadIdx.x & 127);  // float4 index along w (512/4)

    fti_f32x4 xi[4];

#if defined(FTI_HAS_TDM)
    // LDS tile layout: smem[i][hh][w] -> i stride 1024 floats, hh stride 512.
    __shared__ __align__(16) float smem[4 * 2 * 512];

    if (threadIdx.x < 32) {
        // Global start: x[b, j, 2*hb, 0]   (channel i=0 of this j)
        const uint64_t gaddr = (uint64_t)(uintptr_t)x +
            4ull * (((uint64_t)(b * 16 + j) * 512u + (uint64_t)(2 * hb)) * 512u);
        const uint32_t laddr = (uint32_t)(uintptr_t)&smem[0];

        // ---- D# group 0 (CDNA5 ISA) ----
        fti_u32x4 g0 = {};
        g0[0] = 1u;                                      // count=1, flags 0
        g0[1] = laddr;                                   // lds_addr (bytes)
        g0[2] = (uint32_t)(gaddr & 0xFFFFFFFFu);         // global_addr[31:0]
        g0[3] = (uint32_t)((gaddr >> 32) & 0x01FFFFFFu)  // global_addr[56:32]
              | (2u << 30);                              // type = 2 ("image")

        // ---- D# group 1 (§8.4) ----
        // 3-D tile: dim0 = w (512 elems), dim1 = h rows (tile 2, stride 512),
        // dim2 = channel i (tile 4, stride 4*512*512 = 0x100000 elems).
        fti_i32x8 g1 = {};
        g1[0] = (int)(2u << 16);                 // data_size=2 (4B), mask=0
        g1[1] = (int)(512u << 16);               // tensor_dim0[15:0] @63:48
        g1[2] = (int)(512u << 16);               // tensor_dim1[15:0] @95:80
        g1[3] = (int)(512u << 16);               // tile_dim0 @127:112
        g1[4] = (int)((4u << 16) | 2u);          // tile_dim2=4 | tile_dim1=2
        g1[5] = 512;                             // tensor_dim0_stride[31:0]
        g1[6] = 0;                               // stride0 hi | stride1[15:0]
        g1[7] = 16;                              // tensor_dim1_stride >> 16 (0x100000)

        // ---- D# group 2 (§8.5): tensor_dim2 = 4, iterate off ----
        fti_i32x4 g2 = {};
        g2[0] = 4;

        // ---- D# group 3 (§8.6): unused for 3-D ----
        fti_i32x4 g3 = {};

#if FTI_TDM_ARITY == 6
        fti_i32x8 g4 = {};
        __builtin_amdgcn_tensor_load_to_lds(g0, g1, g2, g3, g4, 0);
#else
        __builtin_amdgcn_tensor_load_to_lds(g0, g1, g2, g3, 0);
#endif
#if __has_builtin(__builtin_amdgcn_s_wait_tensorcnt)
        __builtin_amdgcn_s_wait_tensorcnt(0);
#else
        asm volatile("s_wait_tensorcnt 0" ::: "memory");
#endif
    }
    __syncthreads();

    const fti_f32x4* l4 = reinterpret_cast<const fti_f32x4*>(smem);
#pragma unroll
    for (int i = 0; i < 4; ++i)
        xi[i] = l4[i * 256 + hh * 128 + wq];        // ds_load_b128
#else
    // Fallback: direct coalesced b128 global loads (no TDM builtin).
    const fti_f32x4* g4p = reinterpret_cast<const fti_f32x4*>(x);
#pragma unroll
    for (int i = 0; i < 4; ++i) {
        const size_t c = (size_t)(b * 16 + i * 4 + j);
        xi[i] = g4p[(c * 512 + (size_t)(2 * hb + hh)) * 128 + (size_t)wq];
    }
#endif

    // Separable 2x2 synthesis butterfly:
    //   out(2h+a, 2w+c) = (x0*p_a + x1*q_a)*p_c + (x2*p_a + x3*q_a)*q_c
    fti_f32x4 e0, d0, e1, d1;   // row0 even/odd cols, row1 even/odd cols
#pragma unroll
    for (int k = 0; k < 4; ++k) {
        const float a0 = xi[0][k];
        const float a1 = xi[1][k];
        const float a2 = xi[2][k];
        const float a3 = xi[3][k];
        const float u0 = a0 * p0 + a1 * q0;   // row a=0
        const float v0 = a2 * p0 + a3 * q0;
        const float u1 = a0 * p1 + a1 * q1;   // row a=1
        const float v1 = a2 * p1 + a3 * q1;
        e0[k] = u0 * p0 + v0 * q0;            // row0, even col
        d0[k] = u0 * p1 + v0 * q1;            // row0, odd col
        e1[k] = u1 * p0 + v1 * q0;            // row1, even col
        d1[k] = u1 * p1 + v1 * q1;            // row1, odd col
    }

    // Interleave even/odd columns into two contiguous float4s per row.
    const fti_f32x4 r0a = { e0[0], d0[0], e0[1], d0[1] };
    const fti_f32x4 r0b = { e0[2], d0[2], e0[3], d0[3] };
    const fti_f32x4 r1a = { e1[0], d1[0], e1[1], d1[1] };
    const fti_f32x4 r1b = { e1[2], d1[2], e1[3], d1[3] };

    // Output rows 4*hb + 2*hh + {0,1}; columns [8*wq, 8*wq+8).
    // Streaming output: non-temporal stores (TH=NT) so the one-pass result
    // does not displace useful L2 lines.
    const int oh = 4 * hb + 2 * hh;
    fti_f32x4* o4 = reinterpret_cast<fti_f32x4*>(out);
    const size_t rbase =
        (((size_t)(b * 4 + j) * 1024u + (size_t)oh) * 1024u) / 4u + 2u * (size_t)wq;
    __builtin_nontemporal_store(r0a, &o4[rbase + 0]);
    __builtin_nontemporal_store(r0b, &o4[rbase + 1]);
    __builtin_nontemporal_store(r1a, &o4[rbase + 256]);   // next row (+1024 floats)
    __builtin_nontemporal_store(r1b, &o4[rbase + 257]);
}

extern "C" void kernel_launch(void* const* d_in, const int* in_sizes, int n_in,
                              void* d_out, int out_size, void* d_ws, size_t ws_size,
                              hipStream_t stream) {
    (void)in_sizes; (void)n_in; (void)d_ws; (void)ws_size; (void)out_size;
    const float* x  = (const float*)d_in[0];
    const float* w1 = (const float*)d_in[1];
    const float* w2 = (const float*)d_in[2];
    float* out = (float*)d_out;

    // 8 batches * 4 output channels * 256 h-pairs = 8192 blocks of 256 threads.
    dim3 grid(8u * 4u * 256u);
    fti_haar_inv_kernel<<<grid, dim3(256), 0, stream>>>(x, w1, w2, out);
}